// S4Continuous_26328149525104
// MI455X (gfx1250) — compile-verified
//
#include <hip/hip_runtime.h>
#include <hip/hip_bf16.h>

typedef __attribute__((ext_vector_type(2))) float v2f;
typedef __attribute__((ext_vector_type(8))) float v8f;

#define DIM     128
#define SEQL    4096
#define BATCHN  32
#define CHUNK   32
#define NCHUNK  (SEQL / CHUNK)      // 128
#define MATSZ   (DIM * DIM)         // 16384 floats
#define STATE   (BATCHN * DIM)      // 4096 floats

// workspace layout (floats)
#define WS_AT   0                   // A^T, normal row-major (A-operand role in k_power)
#define WS_ATP  (1 * MATSZ)         // A^T, K-paired (B-operand role)
#define WS_BTP  (2 * MATSZ)         // B^T, K-paired
#define WS_CTP  (3 * MATSZ)         // C^T, K-paired
#define WS_MTP  (4 * MATSZ)         // (A^T)^CHUNK, K-paired
#define WS_S    (5 * MATSZ)                    // NCHUNK chunk-local final states
#define WS_HIN  (WS_S + NCHUNK * STATE)        // NCHUNK incoming states

// ---------------------------------------------------------------------------
// fp32 WMMA 16x16x4:  D = A(16x4) * B(4x16) + C   (wave32, exact fp32 path)
// ---------------------------------------------------------------------------
__device__ __forceinline__ v8f wmma4(v2f a, v2f b, v8f c) {
  return __builtin_amdgcn_wmma_f32_16x16x4_f32(
      /*neg_a=*/false, a, /*neg_b=*/false, b,
      /*c_mod=*/(short)0, c, /*reuse_a=*/false, /*reuse_b=*/false);
}

// acc += A[mbase:mbase+16, 0:128] * B[0:128, nbase:nbase+16]
// A: normal row-major (ld=128).  B: K-paired layout, so every fragment
// operand is a single aligned b64 load (no register shuffling).
//   A 16x4 f32 frag: lane l16 = row, half = K-pair select -> v2f at +k
//   B 4x16 f32 frag: lane l16 = col, half = K-pair select -> v2f at pair row
__device__ __forceinline__ v8f mm128p(v8f acc,
                                      const float* __restrict__ Ab, int mbase,
                                      const float* __restrict__ Bp, int nbase,
                                      int l16, int half) {
  const v2f* ap = (const v2f*)(Ab + (size_t)(mbase + l16) * DIM + half * 2);
  const v2f* bp = (const v2f*)(Bp + (size_t)(nbase + l16) * 2 + half * 256);
#pragma unroll
  for (int k = 0; k < DIM; k += 4) {
    v2f a = ap[k >> 1];           // floats {k+2*half, k+2*half+1} of row (mbase+l16)
    v2f b = bp[(k >> 1) * 128];   // pair {k+2*half, k+2*half+1} of col (nbase+l16)
    acc = wmma4(a, b, acc);
  }
  return acc;
}

// store/load a 16x16 f32 C/D tile (VGPR r holds rows mbase+r and mbase+r+8)
__device__ __forceinline__ void stTile(float* dst, int mbase, int nbase,
                                       v8f c, int l16, int half) {
#pragma unroll
  for (int r = 0; r < 8; ++r)
    dst[(mbase + r + half * 8) * DIM + nbase + l16] = c[r];
}
__device__ __forceinline__ void stTileP(float* dstP, int mbase, int nbase,
                                        v8f c, int l16, int half) {
#pragma unroll
  for (int r = 0; r < 8; ++r) {
    int row = mbase + r + half * 8;
    dstP[((row >> 1) * DIM + nbase + l16) * 2 + (row & 1)] = c[r];
  }
}
__device__ __forceinline__ v8f ldTile(const float* src, int mbase, int nbase,
                                      int l16, int half) {
  v8f c;
#pragma unroll
  for (int r = 0; r < 8; ++r)
    c[r] = src[(mbase + r + half * 8) * DIM + nbase + l16];
  return c;
}

// ---------------------------------------------------------------------------
// K0: build A^T (normal) and A^T/B^T/C^T (K-paired) in workspace
// ---------------------------------------------------------------------------
__global__ void k_prep(const float* __restrict__ A,
                       const float* __restrict__ B,
                       const float* __restrict__ C,
                       float* __restrict__ ws) {
  int idx = blockIdx.x * blockDim.x + threadIdx.x;   // 0..4*16384-1
  int mat = idx >> 14;
  int e   = idx & (MATSZ - 1);
  if (mat == 0) {                       // normal transpose: AT[i][j] = A[j][i]
    int i = e >> 7, j = e & 127;
    ws[WS_AT + e] = A[j * DIM + i];
  } else {                              // paired transpose: pairs over k
    int p = e >> 8, n = (e >> 1) & 127, slot = e & 1;
    int k = p * 2 + slot;
    const float* src = (mat == 1) ? A : (mat == 2) ? B : C;
    int off = (mat == 1) ? WS_ATP : (mat == 2) ? WS_BTP : WS_CTP;
    ws[off + e] = src[n * DIM + k];     // T[k][n] = src[n][k]
  }
}

// ---------------------------------------------------------------------------
// K1: MT = (A^T)^CHUNK via 5 LDS-resident WMMA squarings (2^5 = 32).
// Keeps P in both normal (A-role) and paired (B-role) LDS copies.
// ---------------------------------------------------------------------------
__global__ __launch_bounds__(512) void k_power(float* __restrict__ ws) {
  __shared__ __align__(16) float sP[MATSZ];
  __shared__ __align__(16) float sPp[MATSZ];
  int tid = threadIdx.x, lane = tid & 31, wave = tid >> 5;
  int l16 = lane & 15, half = lane >> 4;
  for (int i = tid; i < MATSZ; i += 512) {
    sP[i]  = ws[WS_AT + i];
    sPp[i] = ws[WS_ATP + i];
  }
  __syncthreads();
  for (int it = 0; it < 5; ++it) {
    v8f acc[4];
#pragma unroll
    for (int j = 0; j < 4; ++j) {
      int tile = wave * 4 + j;
      v8f z = {};
      acc[j] = mm128p(z, sP, (tile >> 3) * 16, sPp, (tile & 7) * 16, l16, half);
    }
    __syncthreads();
#pragma unroll
    for (int j = 0; j < 4; ++j) {
      int tile = wave * 4 + j;
      stTile (sP,  (tile >> 3) * 16, (tile & 7) * 16, acc[j], l16, half);
      stTileP(sPp, (tile >> 3) * 16, (tile & 7) * 16, acc[j], l16, half);
    }
    __syncthreads();
  }
  for (int i = tid; i < MATSZ; i += 512) ws[WS_MTP + i] = sPp[i];
}

// ---------------------------------------------------------------------------
// K2: chunk-local recurrence (zero initial state) + partial outputs.
// grid = NCHUNK blocks, 512 threads (16 wave32 = 2x8 WMMA tiles of H).
// LDS: A^Tp (64K) + B^Tp (64K) + H (16K) + Xt (16K) = 160KB.
// ---------------------------------------------------------------------------
__global__ __launch_bounds__(512) void k_chunk(const float* __restrict__ x,
                                               float* __restrict__ y,
                                               float* __restrict__ ws) {
  const float* CTp = ws + WS_CTP;      // global, L2-resident, paired b64 loads
  float* S = ws + WS_S;
  __shared__ __align__(16) float sATp[MATSZ];
  __shared__ __align__(16) float sBTp[MATSZ];
  __shared__ __align__(16) float sH[STATE];
  __shared__ __align__(16) float sX[STATE];

  int tid = threadIdx.x, lane = tid & 31, wave = tid >> 5;
  int l16 = lane & 15, half = lane >> 4;
  int mb = (wave >> 3) * 16, nb = (wave & 7) * 16;
  int c = blockIdx.x;
  int t0 = c * CHUNK;

  for (int i = tid; i < MATSZ; i += 512) {
    sATp[i] = ws[WS_ATP + i];
    sBTp[i] = ws[WS_BTP + i];
  }
  for (int i = tid; i < STATE; i += 512) sH[i] = 0.f;

  // staging: 512 threads x 8 contiguous floats = 32x128 tile
  int e = tid * 8, sb = e >> 7, sd = e & 127;
  __syncthreads();

  for (int t = 0; t < CHUNK; ++t) {
    {  // stage X_t into LDS
      const float4* src = (const float4*)(x + ((long)sb * SEQL + t0 + t) * DIM + sd);
      float4* dst = (float4*)(sX + e);
      dst[0] = src[0];
      dst[1] = src[1];
      if (t + 1 < CHUNK)  // prefetch next timestep (global_prefetch_b8)
        __builtin_prefetch(x + ((long)sb * SEQL + t0 + t + 1) * DIM + sd, 0, 0);
    }
    __syncthreads();

    v8f h = {};
    h = mm128p(h, sH, mb, sATp, nb, l16, half);   // H * A^T
    h = mm128p(h, sX, mb, sBTp, nb, l16, half);   // X_t * B^T
    __syncthreads();                              // all reads of old H done
    stTile(sH, mb, nb, h, l16, half);
    __syncthreads();                              // new H visible

    v8f yv = {};
    yv = mm128p(yv, sH, mb, CTp, nb, l16, half);  // Y_t = H_new * C^T
#pragma unroll
    for (int r = 0; r < 8; ++r) {
      int b = mb + r + half * 8;
      y[((long)b * SEQL + t0 + t) * DIM + nb + l16] = yv[r];
    }
  }
  __syncthreads();
  for (int i = tid; i < STATE; i += 512) S[c * STATE + i] = sH[i];
}

// ---------------------------------------------------------------------------
// K3: sequential carry: Hin[c] = Hin[c-1]*MT + S[c-1]   (single workgroup)
// ---------------------------------------------------------------------------
__global__ __launch_bounds__(512) void k_carry(float* __restrict__ ws) {
  const float* S = ws + WS_S;
  float* Hin = ws + WS_HIN;
  __shared__ __align__(16) float sMTp[MATSZ];
  __shared__ __align__(16) float sH[STATE];
  int tid = threadIdx.x, lane = tid & 31, wave = tid >> 5;
  int l16 = lane & 15, half = lane >> 4;
  int mb = (wave >> 3) * 16, nb = (wave & 7) * 16;

  for (int i = tid; i < MATSZ; i += 512) sMTp[i] = ws[WS_MTP + i];
  for (int i = tid; i < STATE; i += 512) { sH[i] = 0.f; Hin[i] = 0.f; }
  __syncthreads();

  for (int c = 1; c < NCHUNK; ++c) {
    v8f acc = ldTile(S + (c - 1) * STATE, mb, nb, l16, half);
    acc = mm128p(acc, sH, mb, sMTp, nb, l16, half);
    __syncthreads();
    stTile(sH, mb, nb, acc, l16, half);
    stTile(Hin + c * STATE, mb, nb, acc, l16, half);
    __syncthreads();
  }
}

// ---------------------------------------------------------------------------
// K4: fixup — propagate incoming state through each chunk, add C-projection
// ---------------------------------------------------------------------------
__global__ __launch_bounds__(512) void k_fix(float* __restrict__ y,
                                             float* __restrict__ ws) {
  const float* CTp = ws + WS_CTP;
  const float* Hin = ws + WS_HIN;
  __shared__ __align__(16) float sATp[MATSZ];
  __shared__ __align__(16) float sG[STATE];
  int tid = threadIdx.x, lane = tid & 31, wave = tid >> 5;
  int l16 = lane & 15, half = lane >> 4;
  int mb = (wave >> 3) * 16, nb = (wave & 7) * 16;
  int c = blockIdx.x + 1;
  int t0 = c * CHUNK;

  for (int i = tid; i < MATSZ; i += 512) sATp[i] = ws[WS_ATP + i];
  for (int i = tid; i < STATE; i += 512) sG[i] = Hin[c * STATE + i];
  __syncthreads();

  for (int t = 0; t < CHUNK; ++t) {
    v8f g = {};
    g = mm128p(g, sG, mb, sATp, nb, l16, half);   // g <- g * A^T
    __syncthreads();
    stTile(sG, mb, nb, g, l16, half);
    __syncthreads();

    v8f yv;                                       // y += g * C^T
#pragma unroll
    for (int r = 0; r < 8; ++r) {
      int b = mb + r + half * 8;
      yv[r] = y[((long)b * SEQL + t0 + t) * DIM + nb + l16];
    }
    yv = mm128p(yv, sG, mb, CTp, nb, l16, half);
#pragma unroll
    for (int r = 0; r < 8; ++r) {
      int b = mb + r + half * 8;
      y[((long)b * SEQL + t0 + t) * DIM + nb + l16] = yv[r];
    }
  }
}

// ---------------------------------------------------------------------------
extern "C" void kernel_launch(void* const* d_in, const int* in_sizes, int n_in,
                              void* d_out, int out_size, void* d_ws, size_t ws_size,
                              hipStream_t stream) {
  const float* x = (const float*)d_in[0];
  const float* A = (const float*)d_in[1];
  const float* B = (const float*)d_in[2];
  const float* C = (const float*)d_in[3];
  float* y  = (float*)d_out;
  float* ws = (float*)d_ws;

  k_prep  <<<(4 * MATSZ) / 256, 256, 0, stream>>>(A, B, C, ws);
  k_power <<<1, 512, 0, stream>>>(ws);
  k_chunk <<<NCHUNK, 512, 0, stream>>>(x, y, ws);
  k_carry <<<1, 512, 0, stream>>>(ws);
  k_fix   <<<NCHUNK - 1, 512, 0, stream>>>(y, ws);
}